// Projection_77936476553267
// MI455X (gfx1250) — compile-verified
//
#include <hip/hip_runtime.h>

typedef __attribute__((ext_vector_type(2))) float v2f;
typedef __attribute__((ext_vector_type(4))) float v4f;
typedef __attribute__((ext_vector_type(8))) float v8f;

#define WAVES_PER_BLOCK 4
#define THREADS_PER_BLOCK (WAVES_PER_BLOCK * 32)
#define ROWS_PER_WAVE 16
#define DIMS 128

// One wave32 owns a 16x128 tile:
//   stage 1: coalesced global -> LDS (nontemporal b128 loads)
//   stage 2: 32x v_wmma_f32_16x16x4_f32 with B = ones  =>  C[m,*] = sum_k x[m,k]^2
//   stage 3: broadcast per-row sq via shfl, compute s = (sq-1)/(sq+1)
//   stage 4: coalesced scaled stores (stride-32 b32, nontemporal) + s at col 128
__global__ __launch_bounds__(THREADS_PER_BLOCK)
void Projection_77936476553267_kernel(const float* __restrict__ x,
                                      float* __restrict__ out,
                                      int ntiles) {
  __shared__ float tile[WAVES_PER_BLOCK][ROWS_PER_WAVE * DIMS];

  const int lane = threadIdx.x & 31;
  const int wave = threadIdx.x >> 5;
  const int tileIdx = blockIdx.x * WAVES_PER_BLOCK + wave;
  if (tileIdx >= ntiles) return;  // wave-uniform: EXEC stays all-ones for WMMA

  const long row0 = (long)tileIdx * ROWS_PER_WAVE;
  const float* __restrict__ src = x + row0 * DIMS;
  float* __restrict__ lds = &tile[wave][0];

  // ---- Stage 1: global -> LDS, 16 contiguous 512B transactions per wave ----
#pragma unroll
  for (int k = 0; k < 16; ++k) {
    const int off = k * DIMS + lane * 4;  // 16B aligned in both spaces
    v4f v = __builtin_nontemporal_load(reinterpret_cast<const v4f*>(src + off));
    *reinterpret_cast<v4f*>(lds + off) = v;
  }

  // ---- Stage 2: row sums-of-squares via WMMA (B == ones is layout-invariant)
  // A layout (32-bit 16x4): lanes 0-15 supply K=0,1; lanes 16-31 supply K=2,3.
  const int m = lane & 15;
  const int hi = lane >> 4;
  const float* arow = lds + m * DIMS + 2 * hi;

  v8f c = {0.f, 0.f, 0.f, 0.f, 0.f, 0.f, 0.f, 0.f};
  const v2f bones = {1.0f, 1.0f};
#pragma unroll
  for (int j = 0; j < 32; ++j) {
    v2f a = *reinterpret_cast<const v2f*>(arow + 4 * j);  // 8B aligned
    v2f asq;
    asq.x = a.x * a.x;
    asq.y = a.y * a.y;
    // (neg_a, A, neg_b, B, c_mod, C, reuse_a, reuse_b)
    c = __builtin_amdgcn_wmma_f32_16x16x4_f32(false, asq, false, bones,
                                              (short)0, c, false, false);
  }

  // ---- Stage 3 + 4: per row, broadcast sq, scale, store -------------------
  // C layout: VGPR r -> M=r (lanes 0-15), M=r+8 (lanes 16-31).
  float* __restrict__ obase = out + row0 * (DIMS + 1);
#pragma unroll
  for (int r = 0; r < 8; ++r) {
    const float sq_lo = __shfl(c[r], 0, 32);   // row r
    const float sq_hi = __shfl(c[r], 16, 32);  // row r + 8

    {
      const float s = (sq_lo - 1.0f) / (sq_lo + 1.0f);  // == -(1-sq)/(1+sq)
      const float g = 1.0f - s;
      const float* lr = lds + r * DIMS;
      float* orow = obase + (long)r * (DIMS + 1);
#pragma unroll
      for (int q = 0; q < 4; ++q) {
        const int col = lane + 32 * q;  // fully coalesced 128B per store
        __builtin_nontemporal_store(g * lr[col], orow + col);
      }
      if (lane == 0) __builtin_nontemporal_store(s, orow + DIMS);
    }
    {
      const float s = (sq_hi - 1.0f) / (sq_hi + 1.0f);
      const float g = 1.0f - s;
      const float* lr = lds + (r + 8) * DIMS;
      float* orow = obase + (long)(r + 8) * (DIMS + 1);
#pragma unroll
      for (int q = 0; q < 4; ++q) {
        const int col = lane + 32 * q;
        __builtin_nontemporal_store(g * lr[col], orow + col);
      }
      if (lane == 0) __builtin_nontemporal_store(s, orow + DIMS);
    }
  }
}

extern "C" void kernel_launch(void* const* d_in, const int* in_sizes, int n_in,
                              void* d_out, int out_size, void* d_ws, size_t ws_size,
                              hipStream_t stream) {
  const float* x = (const float*)d_in[0];
  float* out = (float*)d_out;

  const long total = (long)in_sizes[0];          // N * 128
  const int rows = (int)(total / DIMS);          // N (2^20: multiple of 16)
  const int ntiles = rows / ROWS_PER_WAVE;
  const int blocks = (ntiles + WAVES_PER_BLOCK - 1) / WAVES_PER_BLOCK;

  Projection_77936476553267_kernel<<<blocks, THREADS_PER_BLOCK, 0, stream>>>(
      x, out, ntiles);
}